// NEGCN_74302934221371
// MI455X (gfx1250) — compile-verified
//
#include <hip/hip_runtime.h>
#include <hip/hip_bf16.h>

// ---------------------------------------------------------------------------
// GCN (node + edge graphs) for MI455X / gfx1250.
// Bandwidth-bound workload: GEMMs (~31 GF total) are negligible next to the
// ~13 GB of gather/scatter-atomic traffic (~0.6 ms @ 23.3 TB/s), so we keep
// full fp32 precision and use V_WMMA_F32_16X16X4_F32 for the dense x@W parts.
// The GEMM epilogue writes both H and AGG_init = H*dinv^2 + bias, removing a
// separate zero-fill pass and the H re-read in the final ReLU pass.
// Workspace requirement: ~1.39 GB of d_ws (carved below).
// ---------------------------------------------------------------------------

typedef __attribute__((ext_vector_type(2))) float v2f;
typedef __attribute__((ext_vector_type(8))) float v8f;

#define GCN_N_NODES 100000
#define GCN_N_EDGES 1600000
#define GCN_N_TRANS 3200000

// ---------------------------------------------------------------------------
// Elementwise fill (degree init)
__global__ void gcn_fill_f32(float* __restrict__ p, float v, long n) {
  long i = (long)blockIdx.x * blockDim.x + threadIdx.x;
  if (i < n) p[i] = v;
}

// In-degree accumulation (deg pre-filled with 1.0 for the self loop)
__global__ void gcn_deg_accum(const int* __restrict__ dst, float* __restrict__ deg, int nE) {
  int e = blockIdx.x * blockDim.x + threadIdx.x;
  if (e < nE) atomicAdd(&deg[dst[e]], 1.0f);
}

// deg -> rsqrt(deg), in place
__global__ void gcn_rsqrt_inplace(float* __restrict__ p, long n) {
  long i = (long)blockIdx.x * blockDim.x + threadIdx.x;
  if (i < n) p[i] = rsqrtf(p[i]);
}

// ---------------------------------------------------------------------------
// H = X @ W and AGG = H*dinv^2 + bias   using V_WMMA_F32_16X16X4_F32.
// One wave32 computes one 16x16 output tile; K-loop in steps of 4.
// A (16x4 f32) layout (ISA 7.12.2): lanes 0-15 hold M=lane / K={k,k+1},
//   lanes 16-31 hold M=lane-16 / K={k+2,k+3}.
// B (4x16 f32): row striped across lanes; v0 = rows {k (low lanes), k+2 (high)},
//   v1 = rows {k+1, k+3}, N = lane&15.
// D (16x16 f32, 8 VGPRs): vgpr r -> M = r + 8*(lane>=16), N = lane&15.
// M is an exact multiple of 16 for all layers (100000, 1600000); the tile
// guard is wave-uniform so EXEC stays all-1s at the WMMA (ISA requirement).
__global__ void gcn_gemm_wmma_f32(const float* __restrict__ X,
                                  const float* __restrict__ W,
                                  const float* __restrict__ dinv,
                                  const float* __restrict__ bias,
                                  float* __restrict__ H,
                                  float* __restrict__ AGG,
                                  int M, int K, int N) {
  const int lane = threadIdx.x & 31;
  const long wave = ((long)blockIdx.x * blockDim.x + threadIdx.x) >> 5;
  const int tilesN = N >> 4;
  const long tiles = (long)(M >> 4) * tilesN;
  if (wave >= tiles) return;  // uniform per wave -> whole wave exits together

  const int tm = (int)(wave / tilesN);
  const int tn = (int)(wave % tilesN);
  const int l = lane & 15;
  const int half = lane >> 4;

  const float* __restrict__ xrow = X + (long)(tm * 16 + l) * K;
  const float* __restrict__ wcol = W + tn * 16 + l;

  v8f acc = {0.f, 0.f, 0.f, 0.f, 0.f, 0.f, 0.f, 0.f};
  for (int k = 0; k < K; k += 4) {
    const int ka = k + (half << 1);
    v2f a;
    a.x = xrow[ka];
    a.y = xrow[ka + 1];
    v2f b;
    b.x = wcol[(long)ka * N];
    b.y = wcol[(long)(ka + 1) * N];
    // 8 args: (neg_a, A, neg_b, B, c_mod, C, reuse_a, reuse_b)
    acc = __builtin_amdgcn_wmma_f32_16x16x4_f32(false, a, false, b,
                                                (short)0, acc, false, false);
  }

  // Epilogue: write raw H (for the scatter gather) and the aggregation
  // initializer AGG = H*dinv^2 + bias (self-loop + bias folded in here).
  const int mbase = tm * 16 + (half << 3);
  const long coff = (long)tn * 16 + l;
  const float bcol = bias[tn * 16 + l];
  float* __restrict__ hp = H + (long)mbase * N + coff;
  float* __restrict__ ap = AGG + (long)mbase * N + coff;
#pragma unroll
  for (int r = 0; r < 8; ++r) {
    const float di = dinv[mbase + r];
    const float h = acc[r];
    hp[(long)r * N] = h;
    ap[(long)r * N] = h * (di * di) + bcol;
  }
}

// ---------------------------------------------------------------------------
// Message scatter: for each edge e and each 4-float chunk of the feature dim,
//   AGG[dst[e]] += H[src[e]] * (dinv[src] * dinv[dst])
// F (out dim) is always a multiple of 4 (32/64/128) and rows are 16B aligned.
// Consecutive lanes cover the chunks of one edge -> coalesced float4 gathers.
__global__ void gcn_scatter_add(const int* __restrict__ src,
                                const int* __restrict__ dst,
                                const float* __restrict__ dinv,
                                const float* __restrict__ H,
                                float* __restrict__ AGG,
                                int nE, int F) {
  const int chunks = F >> 2;
  const long tid = (long)blockIdx.x * blockDim.x + threadIdx.x;
  const long total = (long)nE * chunks;
  if (tid >= total) return;
  const int e = (int)(tid / chunks);
  const int c = (int)(tid % chunks);
  const int s = src[e];
  const int d = dst[e];
  const float norm = dinv[s] * dinv[d];
  const float4 hv = *(const float4*)(H + (long)s * F + (c << 2));
  float* out = AGG + (long)d * F + (c << 2);
  atomicAdd(out + 0, hv.x * norm);
  atomicAdd(out + 1, hv.y * norm);
  atomicAdd(out + 2, hv.z * norm);
  atomicAdd(out + 3, hv.w * norm);
}

// ---------------------------------------------------------------------------
// OUT = relu(AGG), vectorized float4 stream (self-loop/bias already in AGG).
__global__ void gcn_relu4(const float* __restrict__ AGG,
                          float* __restrict__ OUT, long n4) {
  const long i = (long)blockIdx.x * blockDim.x + threadIdx.x;
  if (i >= n4) return;
  float4 v = ((const float4*)AGG)[i];
  v.x = fmaxf(v.x, 0.0f);
  v.y = fmaxf(v.y, 0.0f);
  v.z = fmaxf(v.z, 0.0f);
  v.w = fmaxf(v.w, 0.0f);
  ((float4*)OUT)[i] = v;
}

// ---------------------------------------------------------------------------
static inline int gcn_blocks(long n, int bs) { return (int)((n + bs - 1) / bs); }

extern "C" void kernel_launch(void* const* d_in, const int* in_sizes, int n_in,
                              void* d_out, int out_size, void* d_ws, size_t ws_size,
                              hipStream_t stream) {
  (void)in_sizes; (void)n_in; (void)out_size; (void)ws_size;

  const float* feature_v   = (const float*)d_in[0];
  const int*   edge_index  = (const int*)d_in[1];
  const float* feature_e   = (const float*)d_in[2];
  const int*   trans_index = (const int*)d_in[3];
  const float* Wv1 = (const float*)d_in[4];  const float* bv1 = (const float*)d_in[5];
  const float* We1 = (const float*)d_in[6];  const float* be1 = (const float*)d_in[7];
  const float* Wv2 = (const float*)d_in[8];  const float* bv2 = (const float*)d_in[9];
  const float* We2 = (const float*)d_in[10]; const float* be2 = (const float*)d_in[11];
  const float* Wv3 = (const float*)d_in[12]; const float* bv3 = (const float*)d_in[13];
  const float* We3 = (const float*)d_in[14]; const float* be3 = (const float*)d_in[15];

  const int* sv = edge_index;                // src for node graph
  const int* dv = edge_index + GCN_N_EDGES;  // dst for node graph
  const int* se = trans_index;               // src for edge graph
  const int* de = trans_index + GCN_N_TRANS; // dst for edge graph

  // ---- workspace carve (256B-aligned chunks) -----------------------------
  float* ws = (float*)d_ws;
  size_t off = 0;
  auto carve = [&](size_t nfloats) -> float* {
    float* p = ws + off;
    off += (nfloats + 63) & ~(size_t)63;
    return p;
  };
  float* dinv_v = carve(GCN_N_NODES);
  float* dinv_e = carve(GCN_N_EDGES);
  float* nb0 = carve((size_t)GCN_N_NODES * 128);  // node H
  float* nb1 = carve((size_t)GCN_N_NODES * 128);  // node AGG
  float* nb2 = carve((size_t)GCN_N_NODES * 128);  // node X (activations)
  float* eb0 = carve((size_t)GCN_N_EDGES * 64);   // edge H
  float* eb1 = carve((size_t)GCN_N_EDGES * 64);   // edge AGG
  float* eb2 = carve((size_t)GCN_N_EDGES * 64);   // edge X (activations)

  float* out_fv = (float*)d_out;                            // [100000, 32]
  float* out_fe = (float*)d_out + (size_t)GCN_N_NODES * 32; // [1600000, 32]

  const int BS = 256;

  // ---- degree / rsqrt (once per graph, reused across layers) -------------
  gcn_fill_f32<<<gcn_blocks(GCN_N_NODES, BS), BS, 0, stream>>>(dinv_v, 1.0f, GCN_N_NODES);
  gcn_deg_accum<<<gcn_blocks(GCN_N_EDGES, BS), BS, 0, stream>>>(dv, dinv_v, GCN_N_EDGES);
  gcn_rsqrt_inplace<<<gcn_blocks(GCN_N_NODES, BS), BS, 0, stream>>>(dinv_v, GCN_N_NODES);

  gcn_fill_f32<<<gcn_blocks(GCN_N_EDGES, BS), BS, 0, stream>>>(dinv_e, 1.0f, GCN_N_EDGES);
  gcn_deg_accum<<<gcn_blocks(GCN_N_TRANS, BS), BS, 0, stream>>>(de, dinv_e, GCN_N_TRANS);
  gcn_rsqrt_inplace<<<gcn_blocks(GCN_N_EDGES, BS), BS, 0, stream>>>(dinv_e, GCN_N_EDGES);

  // ---- one GCN layer -----------------------------------------------------
  auto layer = [&](const float* X, const float* W, const float* b,
                   float* H, float* AGG, float* OUT,
                   int M, int K, int N,
                   const int* src, const int* dst, const float* dinv, int nE) {
    const long waves = (long)(M >> 4) * (N >> 4);
    gcn_gemm_wmma_f32<<<gcn_blocks(waves * 32, BS), BS, 0, stream>>>(
        X, W, dinv, b, H, AGG, M, K, N);
    const long scat = (long)nE * (N >> 2);
    gcn_scatter_add<<<gcn_blocks(scat, BS), BS, 0, stream>>>(src, dst, dinv, H, AGG, nE, N);
    const long n4 = ((long)M * N) >> 2;
    gcn_relu4<<<gcn_blocks(n4, BS), BS, 0, stream>>>(AGG, OUT, n4);
  };

  // ---- node tower: 128 -> 128 -> 64 -> 32 --------------------------------
  layer(feature_v, Wv1, bv1, nb0, nb1, nb2,    GCN_N_NODES, 128, 128, sv, dv, dinv_v, GCN_N_EDGES);
  layer(nb2,       Wv2, bv2, nb0, nb1, nb2,    GCN_N_NODES, 128,  64, sv, dv, dinv_v, GCN_N_EDGES);
  layer(nb2,       Wv3, bv3, nb0, nb1, out_fv, GCN_N_NODES,  64,  32, sv, dv, dinv_v, GCN_N_EDGES);

  // ---- edge tower: 32 -> 64 -> 64 -> 32 ----------------------------------
  layer(feature_e, We1, be1, eb0, eb1, eb2,    GCN_N_EDGES, 32, 64, se, de, dinv_e, GCN_N_TRANS);
  layer(eb2,       We2, be2, eb0, eb1, eb2,    GCN_N_EDGES, 64, 64, se, de, dinv_e, GCN_N_TRANS);
  layer(eb2,       We3, be3, eb0, eb1, out_fe, GCN_N_EDGES, 64, 32, se, de, dinv_e, GCN_N_TRANS);
}